// get_loss_4973572129197
// MI455X (gfx1250) — compile-verified
//
#include <hip/hip_runtime.h>
#include <hip/hip_bf16.h>
#include <math.h>

typedef __attribute__((ext_vector_type(2))) float v2f;
typedef __attribute__((ext_vector_type(8))) float v8f;

#define BB   8
#define NN   4096
#define NS   256
#define MM   256
#define NSAMP 2048          // NS*8
#define FINF 3.402823466e+38f

// accumulator indices
#define A_CD1 0   // sum of per-sample min over shape
#define A_CD2 1   // sum of per-shape min over samples
#define A_P2S_SHAPE 2
#define A_P2S_SKEL  3
#define A_RAD 4
#define A_NRM 5
#define A_ND  6
#define A_NS  7
#define A_SKN_SUM 8    // 8..15 per batch
#define A_SKN_SSQ 16   // 16..23 per batch

// raw v_sqrt_f32 / v_exp_f32 (loss-accuracy is fine, avoids the IEEE fixup blob)
static __device__ __forceinline__ float fast_sqrt(float x) {
    return __builtin_amdgcn_sqrtf(x);
}
static __device__ __forceinline__ float fast_exp(float x) {
    return __builtin_amdgcn_exp2f(x * 1.44269504088896f);
}

// ---------------------------------------------------------------- init
__global__ void k_init(float* acc, unsigned* colmin, int ncol) {
    int g = blockIdx.x * blockDim.x + threadIdx.x;
    if (g < 64) acc[g] = 0.0f;
    if (g < ncol) colmin[g] = 0x7F800000u;   // +inf
}

// ---------------------------------------------------------------- d_ss (shape x samples) via WMMA f32 16x16x4
// grid: BB*32 blocks (128 shape rows each), block 256 (8 waves x 16 rows)
// Mins are tracked in SQUARED-distance space; sqrt applied after reduction.
__global__ void k_dss(const float* __restrict__ skel_xyz,
                      const float* __restrict__ skel_radius,
                      const float* __restrict__ shape_all,
                      float* acc, unsigned* colMinSS) {
    __shared__ float4 s_smp[NSAMP];       // sample x,y,z,|.|^2  (32 KB)
    __shared__ unsigned s_col[NSAMP];     // per-column min of squared distance

    const int b = blockIdx.x >> 5;
    const int rowChunk = blockIdx.x & 31;
    const int t = threadIdx.x;

    // precompute the 8 dilated-sphere samples of skel point t
    {
        const float cx = skel_xyz[((size_t)b * NS + t) * 3 + 0];
        const float cy = skel_xyz[((size_t)b * NS + t) * 3 + 1];
        const float cz = skel_xyz[((size_t)b * NS + t) * 3 + 2];
        const float r  = skel_radius[(size_t)b * NS + t];
        const float e = 0.57735027f;
#pragma unroll
        for (int d8 = 0; d8 < 8; ++d8) {
            float sx = cx + r * ((d8 & 4) ? -e : e);
            float sy = cy + r * ((d8 & 2) ? -e : e);
            float sz = cz + r * ((d8 & 1) ? -e : e);
            s_smp[t * 8 + d8] = make_float4(sx, sy, sz, sx * sx + sy * sy + sz * sz);
        }
    }
    for (int i = t; i < NSAMP; i += 256) s_col[i] = 0x7F800000u;
    __syncthreads();

    const int wave = t >> 5, lane = t & 31, half = lane >> 4, lm = lane & 15;
    const int row  = rowChunk * 128 + wave * 16 + lm;        // shape row this lane loads
    const float* sp = shape_all + ((size_t)b * NN + row) * 6;
    const float ax = sp[0], ay = sp[1], az = sp[2];
    const float myRowN = ax * ax + ay * ay + az * az;

    v2f Av;
    Av.x = half ? az : ax;
    Av.y = half ? 0.0f : ay;

    float rn[8];
#pragma unroll
    for (int v = 0; v < 8; ++v) rn[v] = __shfl(myRowN, half * 8 + v, 32);

    float rmin[8];                         // squared-space row mins
#pragma unroll
    for (int v = 0; v < 8; ++v) rmin[v] = FINF;

    for (int c0 = 0; c0 < NSAMP; c0 += 16) {
        const int col = c0 + lm;
        const float4 smp = s_smp[col];
        const float colN = smp.w;

        v2f Bv;
        Bv.x = half ? smp.z : smp.x;
        Bv.y = half ? 0.0f : smp.y;

        v8f C = {};
        C = __builtin_amdgcn_wmma_f32_16x16x4_f32(false, Av, false, Bv,
                                                  (short)0, C, false, false);
        float cmin = FINF;
#pragma unroll
        for (int v = 0; v < 8; ++v) {
            float sq = rn[v] + colN - 2.0f * C[v];
            rmin[v] = fminf(rmin[v], sq);
            cmin = fminf(cmin, sq);
        }
        // both half-lanes hit the same slot; ds_min_u32 combines them
        atomicMin(&s_col[col], __float_as_uint(cmin));
    }

#pragma unroll
    for (int off = 1; off < 16; off <<= 1)
#pragma unroll
        for (int v = 0; v < 8; ++v)
            rmin[v] = fminf(rmin[v], __shfl_xor(rmin[v], off, 32));
    if (lm == 0) {
        float s = 0.0f;
#pragma unroll
        for (int v = 0; v < 8; ++v) s += fast_sqrt(fmaxf(rmin[v], 1e-12f));
        atomicAdd(&acc[A_CD2], s);
    }
    __syncthreads();
    for (int i = t; i < NSAMP; i += 256)
        atomicMin(&colMinSS[(size_t)b * NSAMP + i], s_col[i]);
}

// ---------------------------------------------------------------- d_sc / sph (shape x skel) via WMMA
__global__ void k_dsc(const float* __restrict__ skel_xyz,
                      const float* __restrict__ skel_radius,
                      const float* __restrict__ shape_all,
                      float* acc, unsigned* colMinSC) {
    __shared__ float4 s_c[NS];            // skel x,y,z,|.|^2
    __shared__ float s_r[NS];
    __shared__ unsigned s_col[NS];

    const int b = blockIdx.x >> 5;
    const int rowChunk = blockIdx.x & 31;
    const int t = threadIdx.x;

    {
        const float cx = skel_xyz[((size_t)b * NS + t) * 3 + 0];
        const float cy = skel_xyz[((size_t)b * NS + t) * 3 + 1];
        const float cz = skel_xyz[((size_t)b * NS + t) * 3 + 2];
        s_c[t] = make_float4(cx, cy, cz, cx * cx + cy * cy + cz * cz);
        s_r[t] = skel_radius[(size_t)b * NS + t];
        s_col[t] = 0x7F800000u;
    }
    __syncthreads();

    const int wave = t >> 5, lane = t & 31, half = lane >> 4, lm = lane & 15;
    const int row  = rowChunk * 128 + wave * 16 + lm;
    const float* sp = shape_all + ((size_t)b * NN + row) * 6;
    const float ax = sp[0], ay = sp[1], az = sp[2];
    const float myRowN = ax * ax + ay * ay + az * az;

    v2f Av;
    Av.x = half ? az : ax;
    Av.y = half ? 0.0f : ay;

    float rn[8];
#pragma unroll
    for (int v = 0; v < 8; ++v) rn[v] = __shfl(myRowN, half * 8 + v, 32);

    float rmin[8];
#pragma unroll
    for (int v = 0; v < 8; ++v) rmin[v] = FINF;

    for (int c0 = 0; c0 < NS; c0 += 16) {
        const int col = c0 + lm;
        const float4 c = s_c[col];
        const float r = s_r[col];
        const float colN = c.w;

        v2f Bv;
        Bv.x = half ? c.z : c.x;
        Bv.y = half ? 0.0f : c.y;

        v8f C = {};
        C = __builtin_amdgcn_wmma_f32_16x16x4_f32(false, Av, false, Bv,
                                                  (short)0, C, false, false);
        float cmin = FINF;
#pragma unroll
        for (int v = 0; v < 8; ++v) {
            float sq = rn[v] + colN - 2.0f * C[v];
            float d = fast_sqrt(fmaxf(sq, 1e-12f));
            float sph = fabsf(d - r);
            rmin[v] = fminf(rmin[v], sph);
            cmin = fminf(cmin, sph);
        }
        atomicMin(&s_col[col], __float_as_uint(cmin));
    }

#pragma unroll
    for (int off = 1; off < 16; off <<= 1)
#pragma unroll
        for (int v = 0; v < 8; ++v)
            rmin[v] = fminf(rmin[v], __shfl_xor(rmin[v], off, 32));
    if (lm == 0) {
        float s = 0.0f;
#pragma unroll
        for (int v = 0; v < 8; ++v) s += rmin[v];
        atomicAdd(&acc[A_P2S_SHAPE], s);
    }
    __syncthreads();
    atomicMin(&colMinSC[(size_t)b * NS + t], s_col[t]);
}

// ---------------------------------------------------------------- sum column-min arrays
__global__ void k_summin(const unsigned* __restrict__ src, int n, float* acc,
                         int accIdx, int isSquared) {
    int g = blockIdx.x * blockDim.x + threadIdx.x;
    float v = (g < n) ? __uint_as_float(src[g]) : 0.0f;
    if (isSquared && g < n) v = fast_sqrt(fmaxf(v, 1e-12f));
#pragma unroll
    for (int off = 16; off; off >>= 1) v += __shfl_xor(v, off, 32);
    if ((threadIdx.x & 31) == 0) atomicAdd(&acc[accIdx], v);
}

// ---------------------------------------------------------------- radius + normal-length penalties
__global__ void k_radnd(const float* __restrict__ skel_radius,
                        const float* __restrict__ skel_nori, float* acc) {
    int g = blockIdx.x * blockDim.x + threadIdx.x;   // BB*NS
    float r = skel_radius[g];
    const float nx = skel_nori[(size_t)g * 3 + 0];
    const float ny = skel_nori[(size_t)g * 3 + 1];
    const float nz = skel_nori[(size_t)g * 3 + 2];
    float nrm = fast_sqrt(nx * nx + ny * ny + nz * nz + 1e-12f);
    float nd = fast_exp(-(nrm - 0.3f)) + (nrm < 0.25f ? 5.0f : 0.0f);
#pragma unroll
    for (int off = 16; off; off >>= 1) {
        r += __shfl_xor(r, off, 32);
        nd += __shfl_xor(nd, off, 32);
    }
    if ((threadIdx.x & 31) == 0) {
        atomicAdd(&acc[A_RAD], r);
        atomicAdd(&acc[A_ND], nd);
    }
}

// ---------------------------------------------------------------- loss_normal: top-30 nearest shape pts per skel pt
__global__ void k_normal(const float* __restrict__ skel_xyz,
                         const float* __restrict__ skel_nori,
                         const float* __restrict__ shape_all, float* acc) {
    __shared__ float s_tx[128], s_ty[128], s_tz[128];
    const int b = blockIdx.x, j = threadIdx.x;
    const float cx = skel_xyz[((size_t)b * NS + j) * 3 + 0];
    const float cy = skel_xyz[((size_t)b * NS + j) * 3 + 1];
    const float cz = skel_xyz[((size_t)b * NS + j) * 3 + 2];

    float bd[30];
    int bi[30];
#pragma unroll
    for (int k = 0; k < 30; ++k) { bd[k] = FINF; bi[k] = 0; }
    float wv = FINF; int wp = 0;

    for (int t0 = 0; t0 < NN; t0 += 128) {
        __syncthreads();
        if (j < 128) {
            const float* p = shape_all + ((size_t)b * NN + t0 + j) * 6;
            s_tx[j] = p[0]; s_ty[j] = p[1]; s_tz[j] = p[2];
        }
        __syncthreads();
        for (int i = 0; i < 128; ++i) {
            float dx = cx - s_tx[i], dy = cy - s_ty[i], dz = cz - s_tz[i];
            float dsq = dx * dx + dy * dy + dz * dz;
            if (dsq < wv) {
                bd[wp] = dsq; bi[wp] = t0 + i;
                wv = bd[0]; wp = 0;
                for (int k = 1; k < 30; ++k)
                    if (bd[k] > wv) { wv = bd[k]; wp = k; }
            }
        }
    }
    const float nx = skel_nori[((size_t)b * NS + j) * 3 + 0];
    const float ny = skel_nori[((size_t)b * NS + j) * 3 + 1];
    const float nz = skel_nori[((size_t)b * NS + j) * 3 + 2];
    float s = 0.0f;
    for (int k = 0; k < 30; ++k) {
        const float* nr = shape_all + ((size_t)b * NN + bi[k]) * 6 + 3;
        s += fabsf(nx * nr[0] + ny * nr[1] + nz * nr[2]);
    }
    atomicAdd(&acc[A_NRM], s / 30.0f);
}

// ---------------------------------------------------------------- loss_skn: min dist of combine pts to l3
__global__ void k_skn(const float* __restrict__ skel_xyz,
                      const float* __restrict__ skel_nori,
                      const float* __restrict__ l3_xyz, float* acc) {
    __shared__ float s_lx[MM], s_ly[MM], s_lz[MM];
    const int b = blockIdx.x / 30;                 // 30 blocks per batch
    const int r = (blockIdx.x % 30) * 256 + threadIdx.x;  // 0..7679
    const int t = threadIdx.x;
    s_lx[t] = l3_xyz[((size_t)b * MM + t) * 3 + 0];
    s_ly[t] = l3_xyz[((size_t)b * MM + t) * 3 + 1];
    s_lz[t] = l3_xyz[((size_t)b * MM + t) * 3 + 2];
    __syncthreads();

    const int j = r / 30, k = r % 30;
    const float f = (float)k / 30.0f;
    const float px = skel_xyz[((size_t)b * NS + j) * 3 + 0] + skel_nori[((size_t)b * NS + j) * 3 + 0] * f;
    const float py = skel_xyz[((size_t)b * NS + j) * 3 + 1] + skel_nori[((size_t)b * NS + j) * 3 + 1] * f;
    const float pz = skel_xyz[((size_t)b * NS + j) * 3 + 2] + skel_nori[((size_t)b * NS + j) * 3 + 2] * f;

    float minsq = FINF;
    for (int i = 0; i < MM; ++i) {
        float dx = px - s_lx[i], dy = py - s_ly[i], dz = pz - s_lz[i];
        float sq = dx * dx + dy * dy + dz * dz;
        minsq = fminf(minsq, sq);
    }
    float mind = fast_sqrt(fmaxf(minsq, 1e-12f));
    float sm = mind, sq2 = mind * mind;
#pragma unroll
    for (int off = 16; off; off >>= 1) {
        sm += __shfl_xor(sm, off, 32);
        sq2 += __shfl_xor(sq2, off, 32);
    }
    if ((t & 31) == 0) {
        atomicAdd(&acc[A_SKN_SUM + b], sm);
        atomicAdd(&acc[A_SKN_SSQ + b], sq2);
    }
}

// ---------------------------------------------------------------- loss_ns: 3 nearest skel pts
__global__ void k_ns(const float* __restrict__ skel_xyz,
                     const float* __restrict__ skel_nori, float* acc) {
    __shared__ float s_x[NS], s_y[NS], s_z[NS], s_nx[NS], s_ny[NS], s_nz[NS];
    const int b = blockIdx.x, j = threadIdx.x;
    s_x[j]  = skel_xyz[((size_t)b * NS + j) * 3 + 0];
    s_y[j]  = skel_xyz[((size_t)b * NS + j) * 3 + 1];
    s_z[j]  = skel_xyz[((size_t)b * NS + j) * 3 + 2];
    s_nx[j] = skel_nori[((size_t)b * NS + j) * 3 + 0];
    s_ny[j] = skel_nori[((size_t)b * NS + j) * 3 + 1];
    s_nz[j] = skel_nori[((size_t)b * NS + j) * 3 + 2];
    __syncthreads();

    const float cx = s_x[j], cy = s_y[j], cz = s_z[j];
    float d0 = FINF, d1 = FINF, d2 = FINF;
    int i0 = 0, i1 = 0, i2 = 0;
    for (int i = 0; i < NS; ++i) {
        float dx = cx - s_x[i], dy = cy - s_y[i], dz = cz - s_z[i];
        float dsq = dx * dx + dy * dy + dz * dz;
        if (dsq < d2) {
            if (dsq < d1) {
                d2 = d1; i2 = i1;
                if (dsq < d0) { d1 = d0; i1 = i0; d0 = dsq; i0 = i; }
                else { d1 = dsq; i1 = i; }
            } else { d2 = dsq; i2 = i; }
        }
    }
    const float nx = s_nx[j], ny = s_ny[j], nz = s_nz[j];
    float s = 0.0f;
    int pick[3] = { i0, i1, i2 };
#pragma unroll
    for (int k = 0; k < 3; ++k) {
        int i = pick[k];
        float dx = nx - s_nx[i], dy = ny - s_ny[i], dz = nz - s_nz[i];
        s += fast_sqrt(dx * dx + dy * dy + dz * dz + 1e-12f);
    }
    atomicAdd(&acc[A_NS], s);
}

// ---------------------------------------------------------------- finalize
__global__ void k_final(const float* __restrict__ acc,
                        const int* w0, const int* w1, const int* w2,
                        const int* w4, const int* w5, const int* w6,
                        float* out) {
    if (threadIdx.x != 0 || blockIdx.x != 0) return;
    float ls = acc[A_CD1] / (float)NSAMP + acc[A_CD2] / (float)NN;
    float lp = acc[A_P2S_SHAPE] / (float)NN + acc[A_P2S_SKEL] / (float)NS;
    float lr = -acc[A_RAD] / (float)NS;
    float ln = acc[A_NRM] / (float)BB;
    float total = 0.0f, varsum = 0.0f;
    const float n = (float)(NS * 30);
    for (int b = 0; b < BB; ++b) {
        float s = acc[A_SKN_SUM + b], ss = acc[A_SKN_SSQ + b];
        total += s;
        varsum += (ss - s * s / n) / (n - 1.0f);
    }
    float lskn = 50.0f * total / (float)(BB * NS * 30) + 500.0f * varsum / (float)BB;
    float lnd = acc[A_ND] / (float)(BB * NS);
    float lns = acc[A_NS] / (float)(BB * NS * 3);
    out[0] = (float)(*w0) * ls + (float)(*w1) * lp + (float)(*w2) * lr +
             (float)(*w4) * ln + (float)(*w5) * lskn + (float)(*w6) * lnd +
             0.1f * lns;
}

// ================================================================ host
extern "C" void kernel_launch(void* const* d_in, const int* in_sizes, int n_in,
                              void* d_out, int out_size, void* d_ws, size_t ws_size,
                              hipStream_t stream) {
    const float* skel_xyz    = (const float*)d_in[0];
    const float* skel_radius = (const float*)d_in[1];
    const float* skel_nori   = (const float*)d_in[3];
    const float* l3_xyz      = (const float*)d_in[5];
    const float* shape_all   = (const float*)d_in[7];
    const int* w0 = (const int*)d_in[9];
    const int* w1 = (const int*)d_in[10];
    const int* w2 = (const int*)d_in[11];
    const int* w4 = (const int*)d_in[13];
    const int* w5 = (const int*)d_in[14];
    const int* w6 = (const int*)d_in[15];

    float* acc = (float*)d_ws;
    unsigned* colMinSS = (unsigned*)(acc + 64);          // BB*NSAMP = 16384 (squared mins)
    unsigned* colMinSC = colMinSS + BB * NSAMP;          // BB*NS    = 2048  (sph mins)
    const int ncol = BB * NSAMP + BB * NS;               // 18432 (contiguous)

    k_init<<<(ncol + 255) / 256, 256, 0, stream>>>(acc, colMinSS, ncol);

    k_dss<<<BB * 32, 256, 0, stream>>>(skel_xyz, skel_radius, shape_all, acc, colMinSS);
    k_dsc<<<BB * 32, 256, 0, stream>>>(skel_xyz, skel_radius, shape_all, acc, colMinSC);

    k_summin<<<(BB * NSAMP) / 256, 256, 0, stream>>>(colMinSS, BB * NSAMP, acc, A_CD1, 1);
    k_summin<<<(BB * NS) / 256, 256, 0, stream>>>(colMinSC, BB * NS, acc, A_P2S_SKEL, 0);

    k_radnd<<<BB, 256, 0, stream>>>(skel_radius, skel_nori, acc);
    k_normal<<<BB, 256, 0, stream>>>(skel_xyz, skel_nori, shape_all, acc);
    k_skn<<<BB * 30, 256, 0, stream>>>(skel_xyz, skel_nori, l3_xyz, acc);
    k_ns<<<BB, 256, 0, stream>>>(skel_xyz, skel_nori, acc);

    k_final<<<1, 32, 0, stream>>>(acc, w0, w1, w2, w4, w5, w6, (float*)d_out);
}